// RFInterDAS_53223234732544
// MI455X (gfx1250) — compile-verified
//
#include <hip/hip_runtime.h>

#ifndef __has_builtin
#define __has_builtin(x) 0
#endif

#if __has_builtin(__builtin_amdgcn_global_load_async_to_lds_b128)
#define USE_ASYNC_LDS 1
#else
#define USE_ASYNC_LDS 0
#endif

#define AS1 __attribute__((address_space(1)))
#define AS3 __attribute__((address_space(3)))

typedef __attribute__((ext_vector_type(4))) int v4i;

// Problem constants (match reference setup_inputs()).
constexpr int   A_ANG = 8;
constexpr int   N_ELE = 128;
constexpr int   N_SMP = 2048;
constexpr float C0    = 1540.0f;
constexpr float FS    = 5.0e6f;

constexpr int BLOCK = 256;   // 8 waves (wave32)
constexpr int PIX   = 2;     // pixels per thread
constexpr int TILE  = BLOCK * PIX;

// Stage rf[:, e, 0:S] (8 traces, 64 KB) into one LDS buffer with async b128 copies.
__device__ __forceinline__ void stage_elem(const float* __restrict__ rf, int e,
                                           float* dst, int tid) {
#if USE_ASYNC_LDS
    #pragma unroll
    for (int a = 0; a < A_ANG; ++a) {
        const float* gsrc = rf + (size_t)(a * N_ELE + e) * N_SMP;
        #pragma unroll
        for (int k = 0; k < N_SMP / (BLOCK * 4); ++k) {   // 2 x b128 per thread per trace
            const int i4 = (k * BLOCK + tid) * 4;
            __builtin_amdgcn_global_load_async_to_lds_b128(
                (AS1 v4i*)(gsrc + i4),
                (AS3 v4i*)(dst + a * N_SMP + i4),
                0, 0);
        }
    }
#else
    #pragma unroll
    for (int a = 0; a < A_ANG; ++a) {
        const float4* gsrc = (const float4*)(rf + (size_t)(a * N_ELE + e) * N_SMP);
        float4*       ldst = (float4*)(dst + a * N_SMP);
        #pragma unroll
        for (int k = 0; k < N_SMP / (BLOCK * 4); ++k) {
            const int i4 = k * BLOCK + tid;
            ldst[i4] = gsrc[i4];
        }
    }
#endif
}

__device__ __forceinline__ void wait_async_done() {
#if USE_ASYNC_LDS
#if __has_builtin(__builtin_amdgcn_s_wait_asynccnt)
    __builtin_amdgcn_s_wait_asynccnt(0);
#else
    asm volatile("s_wait_asynccnt 0" ::: "memory");
#endif
#endif
}

__launch_bounds__(BLOCK)
__global__ void das_beamform_kernel(const float* __restrict__ d_tx,
                                    const float* __restrict__ d_rx,
                                    const float* __restrict__ apod,
                                    const float* __restrict__ rf,
                                    const float* __restrict__ t0,
                                    float* __restrict__ out,
                                    int npix) {
    // Double-buffered element traces: 2 * 8 * 2048 * 4B = 128 KB (WGP has 320 KB).
    __shared__ float lds_rf[2][A_ANG * N_SMP];

    const int   tid   = threadIdx.x;
    const int   base  = blockIdx.x * TILE;
    const float scale = FS / C0;
    const float smax  = (float)N_SMP - 1.001f;

    int   pix[PIX];      // true pixel (for guarding stores)
    int   pc[PIX];       // clamped pixel (branchless loads)
    float dtx_s[PIX][A_ANG];   // t0[a]*FS + d_tx[a,pix]*FS/C0
    float acc[PIX][A_ANG];

    #pragma unroll
    for (int p = 0; p < PIX; ++p) {
        pix[p] = base + p * BLOCK + tid;
        pc[p]  = min(pix[p], npix - 1);
        #pragma unroll
        for (int a = 0; a < A_ANG; ++a) {
            float dt    = __builtin_nontemporal_load(&d_tx[a * npix + pc[p]]);
            dtx_s[p][a] = t0[a] * FS + dt * scale;   // t0 uniform -> s_loads
            acc[p][a]   = 0.0f;
        }
    }

    // Prime the pipeline: prefetch element 0 into buffer 0.
    stage_elem(rf, 0, &lds_rf[0][0], tid);

    for (int e = 0; e < N_ELE; ++e) {
        float* cur = &lds_rf[e & 1][0];

        // This wave's outstanding async loads are exactly buf[cur]'s: drain them,
        // then one barrier makes all waves' staged data visible AND guarantees
        // everyone is done reading buf[cur^1] from the previous iteration.
        wait_async_done();
        __syncthreads();

        // Prefetch next element into the other buffer; completes during compute.
        if (e + 1 < N_ELE)
            stage_elem(rf, e + 1, &lds_rf[(e + 1) & 1][0], tid);

        // ---- Gather + lerp + apodized accumulate ----
        #pragma unroll
        for (int p = 0; p < PIX; ++p) {
            // d_rx/apod streamed exactly once over the kernel -> nontemporal
            float drx_s = __builtin_nontemporal_load(&d_rx[e * npix + pc[p]]) * scale;
            float ap    = __builtin_nontemporal_load(&apod[e * npix + pc[p]]);
            #pragma unroll
            for (int a = 0; a < A_ANG; ++a) {
                float s_idx = fminf(fmaxf(dtx_s[p][a] + drx_s, 0.0f), smax);
                int   il    = (int)s_idx;               // s_idx >= 0: trunc == floor
                float frac  = s_idx - (float)il;
                float lo    = cur[a * N_SMP + il];      // ds_load_2addr pair
                float hi    = cur[a * N_SMP + il + 1];
                acc[p][a] += (lo + frac * (hi - lo)) * ap;
            }
        }
    }

    #pragma unroll
    for (int p = 0; p < PIX; ++p) {
        if (pix[p] < npix) {
            #pragma unroll
            for (int a = 0; a < A_ANG; ++a)
                __builtin_nontemporal_store(acc[p][a], &out[a * npix + pix[p]]);
        }
    }
}

extern "C" void kernel_launch(void* const* d_in, const int* in_sizes, int n_in,
                              void* d_out, int out_size, void* d_ws, size_t ws_size,
                              hipStream_t stream) {
    const float* d_tx = (const float*)d_in[0];   // [A, NZ, NX]
    const float* d_rx = (const float*)d_in[1];   // [E, NZ, NX]
    const float* apod = (const float*)d_in[2];   // [E, NZ, NX]
    const float* rf   = (const float*)d_in[3];   // [A, E, S]
    const float* t0   = (const float*)d_in[4];   // [A]
    float*       out  = (float*)d_out;           // [A, NZ, NX]

    const int npix   = in_sizes[0] / A_ANG;      // NZ*NX
    const int blocks = (npix + TILE - 1) / TILE; // 384 for 512x384

    das_beamform_kernel<<<blocks, BLOCK, 0, stream>>>(d_tx, d_rx, apod, rf, t0, out, npix);
}